// SAGEAttackClassifier_84585085928054
// MI455X (gfx1250) — compile-verified
//
#include <hip/hip_runtime.h>
#include <hip/hip_bf16.h>

#define N_NODES 50000
#define N_EDGES 800000
#define DF      128
#define NG      512
#define EPSV    1e-5f

typedef __attribute__((ext_vector_type(2))) float v2f;
typedef __attribute__((ext_vector_type(8))) float v8f;

// ---------------- utility: zero fill ----------------
__global__ __launch_bounds__(256) void zero_f(float* __restrict__ p, int n) {
    int i = blockIdx.x * 256 + threadIdx.x;
    int stride = gridDim.x * 256;
    for (; i < n; i += stride) p[i] = 0.0f;
}

// ---------------- pack [Wl;Wr] (256x128) into WMMA B-fragment order ----------------
// layout: idx = kt*512 + j*64 + lane*2 + q ; k = 4*kt + 2*(lane>>4) + q ; n = 16*j + (lane&15)
__global__ __launch_bounds__(256) void pack_w(const float* __restrict__ wl,
                                              const float* __restrict__ wr,
                                              float* __restrict__ wp) {
    int t = blockIdx.x * 256 + threadIdx.x;
    if (t >= 64 * 8 * 32 * 2) return;
    int q    = t & 1;
    int lane = (t >> 1) & 31;
    int j    = (t >> 6) & 7;
    int kt   = t >> 9;
    int k = 4 * kt + 2 * (lane >> 4) + q;
    int n = 16 * j + (lane & 15);
    wp[t] = (k < DF) ? wl[k * DF + n] : wr[(k - DF) * DF + n];
}

// ---------------- in-degree count ----------------
__global__ __launch_bounds__(256) void edge_count(const int* __restrict__ dst,
                                                  float* __restrict__ cnt) {
    int e = blockIdx.x * 256 + threadIdx.x;
    if (e < N_EDGES) unsafeAtomicAdd(&cnt[dst[e]], 1.0f);
}

// ---------------- edge scatter-add: agg[dst] += feat[src] ----------------
// one wave per edge; lane handles 4 contiguous floats (float4 gather, 4 dword atomics)
__global__ __launch_bounds__(256) void scatter_add(const float* __restrict__ feat,
                                                   const int* __restrict__ src,
                                                   const int* __restrict__ dst,
                                                   float* __restrict__ agg) {
    int e = blockIdx.x * 8 + (threadIdx.x >> 5);
    int lane = threadIdx.x & 31;
    if (e >= N_EDGES) return;
    int s = src[e];
    int d = dst[e];
    const float4 v = *(const float4*)(feat + s * DF + lane * 4);
    float* o = agg + d * DF + lane * 4;
    unsafeAtomicAdd(o + 0, v.x);
    unsafeAtomicAdd(o + 1, v.y);
    unsafeAtomicAdd(o + 2, v.z);
    unsafeAtomicAdd(o + 3, v.w);
}

// ---------------- fused SAGE linear: out = (agg/cnt)@Wl + hin@Wr + b  via WMMA f32 ----------------
// WG = 128 threads = 4 waves; computes rows [16*blk, 16*blk+16), all 128 cols.
// K=256 (cat of normalized agg and hin). A staged in LDS (stride 260 to avoid bank conflicts).
// Safe for out==agg (rows staged to LDS before stores; row tiles are disjoint across WGs).
__global__ __launch_bounds__(128) void sage_gemm(const float* __restrict__ agg,
                                                 const float* __restrict__ hin,
                                                 const float* __restrict__ cnt,
                                                 const float* __restrict__ wpack,
                                                 const float* __restrict__ bias,
                                                 float* __restrict__ out) {
    __shared__ float aCat[16 * 260];
    __shared__ float sInv[16];
    const int tid = threadIdx.x;
    const int m0  = blockIdx.x * 16;

    if (tid < 16) sInv[tid] = 1.0f / fmaxf(cnt[m0 + tid], 1.0f);
    __syncthreads();

    // stage A = [agg/cnt | hin] rows into LDS, coalesced
    for (int idx = tid; idx < 16 * 256; idx += 128) {
        int r = idx >> 8;
        int c = idx & 255;
        float v = (c < DF) ? agg[(m0 + r) * DF + c] * sInv[r]
                           : hin[(m0 + r) * DF + (c - DF)];
        aCat[r * 260 + c] = v;
    }
    __syncthreads();

    const int w    = tid >> 5;
    const int lane = tid & 31;
    const int r    = lane & 15;
    const int koff = 2 * (lane >> 4);

    v8f acc0 = {};
    v8f acc1 = {};
    const float* wp0  = wpack + (2 * w) * 64 + lane * 2;       // n-tile 2w
    const float* wp1  = wpack + (2 * w + 1) * 64 + lane * 2;   // n-tile 2w+1
    const float* aRow = aCat + r * 260 + koff;

    #pragma unroll 4
    for (int kt = 0; kt < 64; ++kt) {
        v2f a  = *(const v2f*)(aRow + 4 * kt);
        v2f b0 = *(const v2f*)(wp0 + kt * 512);
        v2f b1 = *(const v2f*)(wp1 + kt * 512);
        acc0 = __builtin_amdgcn_wmma_f32_16x16x4_f32(false, a, false, b0,
                                                     (short)0, acc0, false, false);
        acc1 = __builtin_amdgcn_wmma_f32_16x16x4_f32(false, a, false, b1,
                                                     (short)0, acc1, false, false);
    }

    const int n0 = 32 * w + (lane & 15);
    const int mo = (lane >> 4) << 3;       // +8 rows for upper half-wave
    const float bb0 = bias[n0];
    const float bb1 = bias[n0 + 16];
    #pragma unroll
    for (int v = 0; v < 8; ++v) {
        int row = m0 + v + mo;
        out[row * DF + n0]      = acc0[v] + bb0;
        out[row * DF + n0 + 16] = acc1[v] + bb1;
    }
}

// ---------------- per-column sum / sum-of-squares ----------------
__global__ __launch_bounds__(128) void col_stats(const float* __restrict__ h,
                                                 float* __restrict__ stats) {
    int col = threadIdx.x;
    int r0 = blockIdx.x * 64;
    float s = 0.0f, sq = 0.0f;
    for (int r = 0; r < 64; ++r) {
        int m = r0 + r;
        if (m < N_NODES) {
            float v = h[m * DF + col];
            s += v;
            sq += v * v;
        }
    }
    unsafeAtomicAdd(&stats[col], s);
    unsafeAtomicAdd(&stats[DF + col], sq);
}

// ---------------- BN (training stats) + ReLU, in place ----------------
__global__ __launch_bounds__(256) void bn_relu(float* __restrict__ h,
                                               const float* __restrict__ stats,
                                               const float* __restrict__ gamma,
                                               const float* __restrict__ beta,
                                               int n) {
    int i = blockIdx.x * 256 + threadIdx.x;
    if (i >= n) return;
    int col = i & (DF - 1);
    const float invN = 1.0f / (float)N_NODES;
    float mu  = stats[col] * invN;
    float var = stats[DF + col] * invN - mu * mu;
    float y = (h[i] - mu) * gamma[col] * rsqrtf(var + EPSV) + beta[col];
    h[i] = fmaxf(y, 0.0f);
}

// ---------------- global mean pool (accumulate) ----------------
__global__ __launch_bounds__(128) void pool_add(const float* __restrict__ h,
                                                const int* __restrict__ batch,
                                                float* __restrict__ pooled,
                                                float* __restrict__ gcnt) {
    int m = blockIdx.x;
    int g = batch[m];
    unsafeAtomicAdd(&pooled[g * DF + threadIdx.x], h[m * DF + threadIdx.x]);
    if (threadIdx.x == 0) unsafeAtomicAdd(&gcnt[g], 1.0f);
}

// ---------------- classifier head: sigmoid((pooled/cnt)@wlin + blin) ----------------
__global__ __launch_bounds__(256) void head(const float* __restrict__ pooled,
                                            const float* __restrict__ gcnt,
                                            const float* __restrict__ wlin,
                                            const float* __restrict__ blin,
                                            float* __restrict__ out) {
    int t = blockIdx.x * 256 + threadIdx.x;
    if (t >= NG * 2) return;
    int g = t >> 1;
    int c = t & 1;
    float inv = 1.0f / fmaxf(gcnt[g], 1.0f);
    float s = 0.0f;
    for (int k = 0; k < DF; ++k) s += pooled[g * DF + k] * wlin[k * 2 + c];
    s = s * inv + blin[c];
    out[t] = 1.0f / (1.0f + expf(-s));
}

// ---------------- launch ----------------
extern "C" void kernel_launch(void* const* d_in, const int* in_sizes, int n_in,
                              void* d_out, int out_size, void* d_ws, size_t ws_size,
                              hipStream_t stream) {
    const float* x      = (const float*)d_in[0];
    const int*   ei     = (const int*)d_in[1];      // [2,E] flat: src then dst
    const int*   batch  = (const int*)d_in[2];
    const float* w1l    = (const float*)d_in[3];
    const float* w1r    = (const float*)d_in[4];
    const float* b1     = (const float*)d_in[5];
    const float* gamma1 = (const float*)d_in[6];
    const float* beta1  = (const float*)d_in[7];
    const float* w2l    = (const float*)d_in[8];
    const float* w2r    = (const float*)d_in[9];
    const float* b2     = (const float*)d_in[10];
    const float* gamma2 = (const float*)d_in[11];
    const float* beta2  = (const float*)d_in[12];
    const float* wlin   = (const float*)d_in[13];
    const float* blin   = (const float*)d_in[14];
    float* out = (float*)d_out;

    const int* src = ei;
    const int* dst = ei + N_EDGES;

    // workspace carve (floats)
    float* ws     = (float*)d_ws;
    float* buf0   = ws;                            // N*128  (agg, later h2)
    float* buf1   = buf0 + N_NODES * DF;           // N*128  (h1)
    float* cnt    = buf1 + N_NODES * DF;           // N
    float* stats  = cnt + N_NODES;                 // 256
    float* pooled = stats + 256;                   // 512*128
    float* gcnt   = pooled + NG * DF;              // 512
    float* wpack1 = gcnt + NG;                     // 32768
    float* wpack2 = wpack1 + 64 * 8 * 32 * 2;      // 32768

    const int smallRegion = N_NODES + 256 + NG * DF + NG;  // cnt..gcnt contiguous

    // init
    zero_f<<<2048, 256, 0, stream>>>(buf0, N_NODES * DF);
    zero_f<<<512, 256, 0, stream>>>(cnt, smallRegion);
    pack_w<<<128, 256, 0, stream>>>(w1l, w1r, wpack1);
    pack_w<<<128, 256, 0, stream>>>(w2l, w2r, wpack2);
    edge_count<<<(N_EDGES + 255) / 256, 256, 0, stream>>>(dst, cnt);

    // ---- layer 1 ----
    scatter_add<<<N_EDGES / 8, 256, 0, stream>>>(x, src, dst, buf0);
    sage_gemm<<<N_NODES / 16, 128, 0, stream>>>(buf0, x, cnt, wpack1, b1, buf1);
    col_stats<<<(N_NODES + 63) / 64, 128, 0, stream>>>(buf1, stats);
    bn_relu<<<(N_NODES * DF) / 256, 256, 0, stream>>>(buf1, stats, gamma1, beta1,
                                                      N_NODES * DF);

    // ---- layer 2 (agg buffer reused; GEMM writes in place over buf0) ----
    zero_f<<<2048, 256, 0, stream>>>(buf0, N_NODES * DF);
    zero_f<<<1, 256, 0, stream>>>(stats, 256);
    scatter_add<<<N_EDGES / 8, 256, 0, stream>>>(buf1, src, dst, buf0);
    sage_gemm<<<N_NODES / 16, 128, 0, stream>>>(buf0, buf1, cnt, wpack2, b2, buf0);
    col_stats<<<(N_NODES + 63) / 64, 128, 0, stream>>>(buf0, stats);
    bn_relu<<<(N_NODES * DF) / 256, 256, 0, stream>>>(buf0, stats, gamma2, beta2,
                                                      N_NODES * DF);

    // ---- pool + head ----
    pool_add<<<N_NODES, 128, 0, stream>>>(buf0, batch, pooled, gcnt);
    head<<<4, 256, 0, stream>>>(pooled, gcnt, wlin, blin, out);
}